// DctLayer_66090956750899
// MI455X (gfx1250) — compile-verified
//
#include <hip/hip_runtime.h>

typedef float v2f __attribute__((ext_vector_type(2)));
typedef float v8f __attribute__((ext_vector_type(8)));

#define WMMA4(A, B, C) \
    __builtin_amdgcn_wmma_f32_16x16x4_f32(false, (A), false, (B), (short)0, (C), false, false)

__global__ __launch_bounds__(256) void dct_corner_kernel(const float* __restrict__ x,
                                                         float* __restrict__ out) {
    constexpr int W = 512;                         // image width (floats)
    constexpr long long PLANE = 512LL * 512LL;     // one channel plane
    constexpr long long OUT_PLANE = 96LL * PLANE;  // one corner's full output
    constexpr int NT = 4;                          // tiles per wave

    __shared__ __align__(16) float lds[8][NT][256];  // 32KB: per-wave, per-tile slice

    const int lane = threadIdx.x & 31;
    const int wave = threadIdx.x >> 5;
    const int grp  = blockIdx.x * 8 + wave;  // 0..24575 groups of 4 row-adjacent tiles

    // 1024 tiles/channel -> 256 groups/channel; 32 tiles/row -> 8 groups/row
    const int chan = grp >> 8;
    const int gg   = grp & 255;
    const int trow = gg >> 3;
    const int tc0  = (gg & 7) << 2;  // first tile column of the group

    const long long base = (long long)chan * PLANE + (long long)(trow * 16) * W + tc0 * 16;

    const int n     = lane & 15;  // column for B/C/D frags; row m for A frags
    const int h     = lane >> 4;  // half-wave select
    const int mbase = h << 3;     // C/D rows mbase..mbase+7 live in this lane
    const int i     = n & 7;

    // ---- projector L = D_low D_low^T (8x8): per lane only 4 distinct values,
    // j = 2h + {0,1,4,5}; block-diagonal gate zeroes half the fragment.
    const float CC = 3.14159265358979323846f / 16.0f;
    const float a1 = CC * (2.0f * (float)i + 1.0f);
    const float ci1 = __cosf(a1), ci2 = __cosf(2.0f * a1), ci3 = __cosf(3.0f * a1);
    float Lv[4];
#pragma unroll
    for (int s = 0; s < 4; ++s) {
        const int j = 2 * h + ((s >> 1) << 2) + (s & 1);
        const float b1 = CC * (2.0f * (float)j + 1.0f);
        Lv[s] = 0.125f + 0.25f * (ci1 * __cosf(b1) + ci2 * __cosf(2.0f * b1) +
                                  ci3 * __cosf(3.0f * b1));
    }
    // Lf[2q+v] = Lbig[n][4q+2h+v] ; serves as both A-frag and B-frag (Lbig symmetric)
    float Lf[8];
#pragma unroll
    for (int q = 0; q < 4; ++q)
#pragma unroll
        for (int v = 0; v < 2; ++v) {
            const bool nz = (n < 8) ? (q < 2) : (q >= 2);
            Lf[2 * q + v] = nz ? Lv[((q & 1) << 1) + v] : 0.0f;
        }
    const v2f La0 = {Lf[0], Lf[1]}, La1 = {Lf[2], Lf[3]};
    const v2f La2 = {Lf[4], Lf[5]}, La3 = {Lf[6], Lf[7]};

    // ---- hoisted base pointers; all per-tile/per-element deltas are immediates
    const float* pd = x + base + (long long)mbase * W + n;      // dT stream
    const float* pa = x + base + (long long)n * W + 2 * h;      // aT stream (row m=n)
    const float* pb = x + base + (long long)(2 * h) * W + n;    // bT stream
    float* o0 = out + base + (long long)mbase * W + n;          // corner 0
    float* o1 = o0 + OUT_PLANE;
    float* o2 = o1 + OUT_PLANE;
    float* o3 = o2 + OUT_PLANE;
    float* wsb = &lds[wave][0][0] + n * 16 + mbase;             // LDS store base
    const float* wlb = &lds[wave][0][0] + n * 16 + 2 * h;       // LDS load base

#pragma unroll
    for (int tt = 0; tt < NT; ++tt) {
        const int co = tt * 16;  // compile-time column offset (floats)

        // ---- tile T in C/D layout
        v8f dT;
#pragma unroll
        for (int r = 0; r < 8; ++r) dT[r] = pd[r * W + co];

        // ---- T as A-frag (T[m][4q+2h+v]) and B-frag (T[4q+2h+v][n])
        float aT[8], bT[8];
#pragma unroll
        for (int q = 0; q < 4; ++q)
#pragma unroll
            for (int v = 0; v < 2; ++v) {
                aT[2 * q + v] = pa[4 * q + v + co];
                bT[2 * q + v] = pb[(4 * q + v) * W + co];
            }

        // ---- Y = Lbig * T
        v8f accY = {0.f, 0.f, 0.f, 0.f, 0.f, 0.f, 0.f, 0.f};
        accY = WMMA4(La0, ((v2f){bT[0], bT[1]}), accY);
        accY = WMMA4(La1, ((v2f){bT[2], bT[3]}), accY);
        accY = WMMA4(La2, ((v2f){bT[4], bT[5]}), accY);
        accY = WMMA4(La3, ((v2f){bT[6], bT[7]}), accY);

        // ---- Z = T * Lbig
        v8f accZ = {0.f, 0.f, 0.f, 0.f, 0.f, 0.f, 0.f, 0.f};
        accZ = WMMA4(((v2f){aT[0], aT[1]}), La0, accZ);
        accZ = WMMA4(((v2f){aT[2], aT[3]}), La1, accZ);
        accZ = WMMA4(((v2f){aT[4], aT[5]}), La2, accZ);
        accZ = WMMA4(((v2f){aT[6], aT[7]}), La3, accZ);

        // ---- route Y through this tile's LDS slice (column-major) to build B-frag
        {
            float4* w4 = (float4*)(wsb + tt * 256);  // 32B aligned
            w4[0] = make_float4(accY[0], accY[1], accY[2], accY[3]);
            w4[1] = make_float4(accY[4], accY[5], accY[6], accY[7]);
        }
        asm volatile("s_wait_dscnt 0" ::: "memory");  // per-wave store->load ordering

        float bY[8];
#pragma unroll
        for (int q = 0; q < 4; ++q) {
            bY[2 * q]     = wlb[tt * 256 + 4 * q];      // Y[4q+2h][n]
            bY[2 * q + 1] = wlb[tt * 256 + 4 * q + 1];  // Y[4q+2h+1][n]
        }

        // ---- W = Lbig * Y = Lbig * Lbig * T
        v8f accW = {0.f, 0.f, 0.f, 0.f, 0.f, 0.f, 0.f, 0.f};
        accW = WMMA4(La0, ((v2f){bY[0], bY[1]}), accW);
        accW = WMMA4(La1, ((v2f){bY[2], bY[3]}), accW);
        accW = WMMA4(La2, ((v2f){bY[4], bY[5]}), accW);
        accW = WMMA4(La3, ((v2f){bY[6], bY[7]}), accW);

        // ---- corners: LLB, LB-LLB, BL-LLB, B-LB-BL+LLB ; streaming NT stores
#pragma unroll
        for (int r = 0; r < 8; ++r) {
            const float w = accW[r], y = accY[r], z = accZ[r];
            __builtin_nontemporal_store(w, o0 + r * W + co);
            __builtin_nontemporal_store(y - w, o1 + r * W + co);
            __builtin_nontemporal_store(z - w, o2 + r * W + co);
            __builtin_nontemporal_store(dT[r] - y - z + w, o3 + r * W + co);
        }
    }
}

extern "C" void kernel_launch(void* const* d_in, const int* in_sizes, int n_in,
                              void* d_out, int out_size, void* d_ws, size_t ws_size,
                              hipStream_t stream) {
    (void)n_in; (void)out_size; (void)d_ws; (void)ws_size;
    const float* x = (const float*)d_in[0];
    float* out = (float*)d_out;

    // 96 planes * 1024 tiles = 98304 tiles; 4 tiles/wave, 8 waves/block -> 3072 blocks
    const int nchan  = in_sizes[0] / (512 * 512);  // = 96
    const int tiles  = nchan * 32 * 32;            // = 98304
    const int blocks = tiles / (8 * 4);            // = 3072

    dct_corner_kernel<<<blocks, 256, 0, stream>>>(x, out);
}